// EBParameterEstimator_23768349016276
// MI455X (gfx1250) — compile-verified
//
#include <hip/hip_runtime.h>

// Problem constants (from reference)
#define BB 4
#define NN 20000
#define EE 160000
#define FF 16
#define HH 3
#define CC 32
#define HC 96           // H*C
#define ROWS (BB*NN)    // 80000, row r = n*B + b

typedef float v2f __attribute__((ext_vector_type(2)));
typedef float v8f __attribute__((ext_vector_type(8)));

// ---------- helpers: order-preserving float<->uint keys for atomic max ----------
__device__ __forceinline__ unsigned f2key(float f) {
  unsigned b = __float_as_uint(f);
  return (b & 0x80000000u) ? ~b : (b | 0x80000000u);
}
__device__ __forceinline__ float key2f(unsigned k) {
  unsigned b = (k & 0x80000000u) ? (k & 0x7FFFFFFFu) : ~k;
  return __uint_as_float(b);
}

// ---------------- K0: zero the accumulation regions ----------------
__global__ void zero_ws(float* p, long n) {
  long i = (long)blockIdx.x * blockDim.x + threadIdx.x;
  long stride = (long)gridDim.x * blockDim.x;
  for (; i < n; i += stride) p[i] = 0.0f;
}

// ---------------- K1: fl/fr = x @ Wl / x @ Wr via V_WMMA_F32_16X16X4_F32 ----------------
// One wave handles one 16-row M tile (rows r = n*B+b) x all 96 cols, for one W.
// grid = 2500 blocks x 128 threads = 10000 waves = 5000 tiles * 2 matrices.
__global__ void transform_wmma(const float* __restrict__ x,
                               const float* __restrict__ Wl,
                               const float* __restrict__ Wr,
                               float* __restrict__ fl,
                               float* __restrict__ fr) {
  const int lane   = threadIdx.x & 31;
  const int gwave  = blockIdx.x * (blockDim.x >> 5) + (threadIdx.x >> 5);
  const int mtile  = gwave >> 1;
  const int which  = gwave & 1;
  const float* __restrict__ W = which ? Wr : Wl;   // [F][96] row-major
  float* __restrict__ out     = which ? fr : fl;   // [ROWS][96]

  const int half = lane >> 4;      // 0: K even pair base 0, 1: base 2
  const int lrow = lane & 15;
  const int row0 = mtile * 16;
  const int r    = row0 + lrow;    // r = n*B + b
  const int n    = r >> 2;
  const int b    = r & 3;
  const float* __restrict__ xrow = x + ((size_t)b * NN + n) * FF;

  // A operand, 4 k-steps: lanes 0-15 hold K={4k,4k+1}, lanes 16-31 hold K={4k+2,4k+3}
  v2f a[4];
#pragma unroll
  for (int kk = 0; kk < 4; ++kk) {
    const int f = 4 * kk + 2 * half;
    a[kk].x = xrow[f];
    a[kk].y = xrow[f + 1];
  }

#pragma unroll
  for (int nt = 0; nt < 6; ++nt) {           // 6 * 16 = 96 output cols
    const int col = nt * 16 + lrow;
    v8f acc = {0.f, 0.f, 0.f, 0.f, 0.f, 0.f, 0.f, 0.f};
#pragma unroll
    for (int kk = 0; kk < 4; ++kk) {
      const int k = 4 * kk + 2 * half;
      v2f bm;
      bm.x = W[(size_t)k * HC + col];
      bm.y = W[(size_t)(k + 1) * HC + col];
      acc = __builtin_amdgcn_wmma_f32_16x16x4_f32(
          /*neg_a=*/false, a[kk], /*neg_b=*/false, bm,
          /*c_mod=*/(short)0, acc, /*reuse_a=*/false, /*reuse_b=*/false);
    }
#pragma unroll
    for (int i = 0; i < 8; ++i) {
      const int orow = row0 + i + 8 * half;      // C/D layout: VGPR i -> M=i / M=i+8
      out[(size_t)orow * HC + nt * 16 + lrow] = acc[i];
    }
  }
}

// ---------------- K2: per-edge GATv2 scores + segment max (atomic key max) ----------------
__global__ void edge_scores(const int* __restrict__ ei,
                            const float* __restrict__ fl,
                            const float* __restrict__ fr,
                            const float* __restrict__ attn,
                            float* __restrict__ score_ws,
                            unsigned* __restrict__ smax_key) {
  const int t = blockIdx.x * blockDim.x + threadIdx.x;  // over E*B*H
  if (t >= EE * BB * HH) return;
  const int h = t % HH;
  const int tb = t / HH;
  const int b = tb % BB;
  const int e = tb / BB;
  const int s = ei[e];
  const int d = ei[EE + e];
  const float* __restrict__ flr = fl + ((size_t)(d * BB + b)) * HC + h * CC;
  const float* __restrict__ frr = fr + ((size_t)(s * BB + b)) * HC + h * CC;
  const float* __restrict__ at  = attn + h * CC;
  float sc = 0.f;
#pragma unroll 8
  for (int c = 0; c < CC; ++c) {
    float v = flr[c] + frr[c];
    v = (v > 0.f) ? v : 0.2f * v;            // LeakyReLU(0.2)
    sc += v * at[c];
  }
  score_ws[t] = sc;
  atomicMax(&smax_key[(d * BB + b) * HH + h], f2key(sc));
}

// ---------------- K3: exp, denom and UNNORMALIZED weighted scatter-add ----------------
__global__ void edge_accum(const int* __restrict__ ei,
                           const float* __restrict__ fr,
                           const float* __restrict__ score_ws,
                           const unsigned* __restrict__ smax_key,
                           float* __restrict__ denom,
                           float* __restrict__ aggun) {
  const int t = blockIdx.x * blockDim.x + threadIdx.x;
  if (t >= EE * BB * HH) return;
  const int h = t % HH;
  const int tb = t / HH;
  const int b = tb % BB;
  const int e = tb / BB;
  const int s = ei[e];
  const int d = ei[EE + e];
  const float smax = key2f(smax_key[(d * BB + b) * HH + h]);
  const float ex = __expf(score_ws[t] - smax);
  atomicAdd(&denom[(d * BB + b) * HH + h], ex);
  const float* __restrict__ frr = fr + ((size_t)(s * BB + b)) * HC + h * CC;
  float* __restrict__ ag = aggun + ((size_t)(d * BB + b)) * HC + h * CC;
#pragma unroll 8
  for (int c = 0; c < CC; ++c) atomicAdd(&ag[c], ex * frr[c]);
}

// ---------------- K4: h = agg/denom + bias, global max-pool over nodes ----------------
// block = 384 threads (one per (b,hc) column); each block scans a node chunk,
// reads are contiguous 1536B per node; one atomicMax key per column per block.
__global__ void pool_max(const float* __restrict__ aggun,
                         const float* __restrict__ denom,
                         const float* __restrict__ bias,
                         unsigned* __restrict__ pooled_key) {
  const int j = threadIdx.x;        // 0..383 = b*96 + hc
  const int b = j / HC;
  const int hc = j % HC;
  const int h = hc >> 5;
  const int npb = (NN + gridDim.x - 1) / gridDim.x;
  int n0 = blockIdx.x * npb;
  int n1 = n0 + npb; if (n1 > NN) n1 = NN;
  const float bi = bias[hc];
  float m = -3.402823466e38f;
  for (int n = n0; n < n1; ++n) {
    const float dn = denom[(n * BB + b) * HH + h];
    const float v = aggun[(size_t)n * (BB * HC) + j] / fmaxf(dn, 1e-16f) + bi;
    m = fmaxf(m, v);
  }
  if (n1 > n0) atomicMax(&pooled_key[j], f2key(m));
}

// ---------------- K5: MLP head (single block) ----------------
__global__ void mlp_head(const unsigned* __restrict__ pooled_key,
                         const float* __restrict__ W1, const float* __restrict__ b1,
                         const float* __restrict__ W2, const float* __restrict__ b2,
                         float* __restrict__ out) {
  __shared__ float pooled[BB * HC];
  __shared__ float z[BB * 128];
  const int t = threadIdx.x;   // 128 threads
  for (int i = t; i < BB * HC; i += 128) pooled[i] = key2f(pooled_key[i]);
  __syncthreads();
  for (int idx = t; idx < BB * 128; idx += 128) {
    const int b = idx >> 7, k = idx & 127;
    float s = b1[k];
#pragma unroll 8
    for (int j = 0; j < HC; ++j) s += pooled[b * HC + j] * W1[j * 128 + k];
    z[idx] = s * (1.f / (1.f + __expf(-s)));   // swish
  }
  __syncthreads();
  for (int idx = t; idx < BB * 7; idx += 128) {
    const int b = idx / 7, k = idx % 7;
    float s = b2[k];
#pragma unroll 8
    for (int j = 0; j < 128; ++j) s += z[b * 128 + j] * W2[j * 7 + k];
    out[b * 7 + k] = s;
  }
}

// ---------------- launcher ----------------
extern "C" void kernel_launch(void* const* d_in, const int* in_sizes, int n_in,
                              void* d_out, int out_size, void* d_ws, size_t ws_size,
                              hipStream_t stream) {
  const float* x    = (const float*)d_in[0];
  const int*   ei   = (const int*)  d_in[1];
  const float* Wl   = (const float*)d_in[2];
  const float* Wr   = (const float*)d_in[3];
  const float* attn = (const float*)d_in[4];
  const float* bias = (const float*)d_in[5];
  const float* W1   = (const float*)d_in[6];
  const float* b1   = (const float*)d_in[7];
  const float* W2   = (const float*)d_in[8];
  const float* b2   = (const float*)d_in[9];
  float* out = (float*)d_out;

  // workspace layout (float words)
  float* ws = (float*)d_ws;
  const size_t n_fl   = (size_t)ROWS * HC;        // 7,680,000
  const size_t n_sc   = (size_t)EE * BB * HH;     // 1,920,000
  const size_t n_seg  = (size_t)NN * BB * HH;     //   240,000
  float*    fl         = ws;
  float*    fr         = fl + n_fl;
  float*    score_ws   = fr + n_fl;
  unsigned* smax_key   = (unsigned*)(score_ws + n_sc);
  float*    denom      = (float*)(smax_key + n_seg);
  float*    aggun      = denom + n_seg;
  unsigned* pooled_key = (unsigned*)(aggun + n_fl);
  const long n_zero = (long)(n_seg /*smax*/ + n_seg /*denom*/ + n_fl /*aggun*/ + BB * HC);

  // K0: zero smax_key..pooled_key (contiguous)
  zero_ws<<<2048, 256, 0, stream>>>((float*)smax_key, n_zero);

  // K1: WMMA transform. 5000 M-tiles * 2 matrices = 10000 waves.
  transform_wmma<<<2500, 128, 0, stream>>>(x, Wl, Wr, fl, fr);

  // K2/K3: edge phase, one thread per (e,b,h)
  const int nebh = EE * BB * HH;
  edge_scores<<<(nebh + 255) / 256, 256, 0, stream>>>(ei, fl, fr, attn, score_ws, smax_key);
  edge_accum <<<(nebh + 255) / 256, 256, 0, stream>>>(ei, fr, score_ws, smax_key, denom, aggun);

  // K4: node finalize + global max pool
  pool_max<<<100, BB * HC, 0, stream>>>(aggun, denom, bias, pooled_key);

  // K5: MLP head
  mlp_head<<<1, 128, 0, stream>>>(pooled_key, W1, b1, W2, b2, out);
}